// LocallyConnected2D_77309411518
// MI455X (gfx1250) — compile-verified
//
#include <hip/hip_runtime.h>

typedef __attribute__((ext_vector_type(2))) float v2f;
typedef __attribute__((ext_vector_type(8))) float v8f;

#define KH     3
#define KW     3
#define C_IN   32
#define H_IN   64
#define W_IN   64
#define BATCH  16
#define F_OUT  64
#define K_DIM  (KH * KW * C_IN)      // 288
#define OUT_H  62
#define OUT_W  62
#define P_DIM  (OUT_H * OUT_W)       // 3844
#define LDS_STRIDE 292               // 288 + pad: m*292 dwords -> conflict-free b64 reads

__global__ __launch_bounds__(128) void lc2d_wmma_kernel(
    const float* __restrict__ x,      // (16, 64, 64, 32)
    const float* __restrict__ kern,   // (P, 288, 64)
    const float* __restrict__ bias,   // (P, 64)
    float* __restrict__ out)          // (16, 62, 62, 64)
{
    __shared__ float a_lds[BATCH * LDS_STRIDE];

    const int p  = blockIdx.x;        // spatial position
    const int oh = p / OUT_W;
    const int ow = p - oh * OUT_W;

    const int tid  = threadIdx.x;     // 0..127
    const int wave = tid >> 5;        // 0..3 -> which 16-col slice of F
    const int lane = tid & 31;

    // ---- Stage A (im2col patches for this position) into LDS ----
    // A[m][k], k = kh*96 + kw*32 + c ; value = x[m][oh+kh][ow+kw][c]
    // 128 threads: thread -> (m = tid/8, c4 = (tid%8)*4), float4 per (kh,kw).
    {
        const int lm = tid >> 3;           // batch row 0..15
        const int lc = (tid & 7) << 2;     // channel 0,4,...,28
        #pragma unroll
        for (int kh = 0; kh < KH; ++kh) {
            #pragma unroll
            for (int kw = 0; kw < KW; ++kw) {
                const float4 v = *(const float4*)(
                    x + ((((size_t)lm * H_IN + (oh + kh)) * W_IN + (ow + kw)) * C_IN + lc));
                *(float4*)(&a_lds[lm * LDS_STRIDE + (kh * KW + kw) * C_IN + lc]) = v;
            }
        }
    }
    __syncthreads();

    // ---- Fragment bases per ISA 16x16x4 f32 layout ----
    // A (16x4): lanes 0-15 -> M=lane, VGPR0=K0,VGPR1=K1 ; lanes 16-31 -> K2,K3
    // B (4x16): lanes 0-15 -> N=lane, rows K0/K1       ; lanes 16-31 -> rows K2/K3
    const int mn    = lane & 15;            // M for A-frag, N for B-frag
    const int khalf = (lane >> 4) << 1;     // 0 or 2

    const float* __restrict__ Arow = a_lds + mn * LDS_STRIDE + khalf;
    const float* __restrict__ Bp   = kern + (size_t)p * (K_DIM * F_OUT)
                                          + wave * 16 + mn;   // + k*F_OUT per row

    v8f acc = {};

    #pragma unroll 8
    for (int k = 0; k < K_DIM; k += 4) {
        // A fragment: two consecutive K values -> single aligned 8B LDS read
        v2f a = *(const v2f*)(Arow + k);
        // B fragment: two weight-row elements (coalesced 64B per half-wave)
        v2f b;
        b.x = Bp[(size_t)(k + khalf)     * F_OUT];
        b.y = Bp[(size_t)(k + khalf + 1) * F_OUT];
        // D = A(16x4) x B(4x16) + C
        acc = __builtin_amdgcn_wmma_f32_16x16x4_f32(
            /*neg_a=*/false, a, /*neg_b=*/false, b,
            /*c_mod=*/(short)0, acc, /*reuse_a=*/false, /*reuse_b=*/false);
    }

    // ---- bias + store ----
    // C/D layout: VGPR v, lanes 0-15 -> M=v ; lanes 16-31 -> M=v+8 ; N=lane&15
    const float bval  = bias[(size_t)p * F_OUT + wave * 16 + mn];
    const int   mbase = (lane >> 4) << 3;   // 0 or 8
    float* __restrict__ outp = out + (size_t)p * F_OUT + wave * 16 + mn;

    #pragma unroll
    for (int v = 0; v < 8; ++v) {
        outp[(size_t)(mbase + v) * (P_DIM * F_OUT)] = acc[v] + bval;
    }
}

extern "C" void kernel_launch(void* const* d_in, const int* in_sizes, int n_in,
                              void* d_out, int out_size, void* d_ws, size_t ws_size,
                              hipStream_t stream) {
    const float* x    = (const float*)d_in[0];
    const float* kern = (const float*)d_in[1];
    const float* bias = (const float*)d_in[2];
    float* out        = (float*)d_out;

    dim3 grid(P_DIM);   // 3844 positions
    dim3 block(128);    // 4 waves: one 16x16 output tile each
    hipLaunchKernelGGL(lc2d_wmma_kernel, grid, block, 0, stream, x, kern, bias, out);
}